// InformationForagingModel_11184094838942
// MI455X (gfx1250) — compile-verified
//
#include <hip/hip_runtime.h>
#include <stdint.h>

// ---------------- problem constants (from reference) ----------------
#define TT      1024          // timesteps
#define FF      19            // feature dim (16 one-hot + partner + time + pid)
#define NTT     16            // tiles
#define OFFV    (-4.0f)
#define TMAXI   (1.0f / 10.0f)

// ---------------- pipelining constants ----------------
#define SS        64                      // steps per chunk
#define NCHUNK    (TT / SS)               // 16 chunks
#define RECB      (FF * 4)                // 76 bytes per step record
#define TRAJ_IN   (SS * RECB)             // 4864 B per trajectory per chunk
#define WAVE_IN   (2 * TRAJ_IN)           // 9728 B per wave per chunk (2 traj)
#define TRAJ_OUT  (SS * NTT * 4)          // 4096 B per trajectory per chunk
#define WAVE_OUT  (2 * TRAJ_OUT)          // 8192 B
#define IN_UNITS  (WAVE_IN / 16)          // 608 b128 units -> 19 per lane (exact)
#define OUT_UNITS (WAVE_OUT / 16)         // 512 b128 units -> 16 per lane (exact)
#define LDS_BYTES (2 * WAVE_IN + WAVE_OUT) // 27648 B

// ---------------- CDNA5 async global<->LDS primitives ----------------
__device__ __forceinline__ void async_load_b128(unsigned lds_addr, unsigned long long gaddr) {
    // DMA 16B global -> LDS, tracked by ASYNCcnt (CDNA5 ISA op 98)
    asm volatile("global_load_async_to_lds_b128 %0, %1, off"
                 :: "v"(lds_addr), "v"(gaddr) : "memory");
}
__device__ __forceinline__ void async_store_b128(unsigned long long gaddr, unsigned lds_addr) {
    // DMA 16B LDS -> global, tracked by ASYNCcnt (op 102)
    asm volatile("global_store_async_from_lds_b128 %0, %1, off"
                 :: "v"(gaddr), "v"(lds_addr) : "memory");
}
__device__ __forceinline__ void wait_async0() {
    asm volatile("s_wait_asynccnt 0" ::: "memory");
}
__device__ __forceinline__ void wait_ds0() {
    asm volatile("s_wait_dscnt 0" ::: "memory");
}

// One wave (=one workgroup) of 32 lanes handles 2 trajectories:
//   lanes 0..15  -> tiles 0..15 of trajectory (2*blockIdx.x + 0)
//   lanes 16..31 -> tiles 0..15 of trajectory (2*blockIdx.x + 1)
extern "C" __global__ __launch_bounds__(32)
void ifm_scan_kernel(const float* __restrict__ in,     // (B,T,1,F)
                     const float* __restrict__ p_nvs_g,// (NP,1)
                     const float* __restrict__ p_nvp_g,
                     const float* __restrict__ p_vbs_g,
                     const float* __restrict__ p_vbp_g,
                     const float* __restrict__ p_mp_g,
                     float* __restrict__ out)          // (B,T,1,NT)
{
    __shared__ char smem[LDS_BYTES];

    const int lane = threadIdx.x;           // 0..31
    const int traj = lane >> 4;             // 0/1: which trajectory this lane computes
    const int tile = lane & 15;             // tile index for this lane
    const long long b = 2LL * blockIdx.x + traj;

    // LDS byte offsets (low 32 bits of the generic shared pointer == LDS address)
    const unsigned sbase    = (unsigned)(uintptr_t)(&smem[0]);
    const unsigned in_lds0  = sbase;                 // input buffer 0 (WAVE_IN)
    const unsigned in_lds1  = sbase + WAVE_IN;       // input buffer 1 (WAVE_IN)
    const unsigned out_ldsb = sbase + 2 * WAVE_IN;   // output staging (WAVE_OUT)
    float* const out_ptr    = (float*)(smem + 2 * WAVE_IN);

    // Global base addresses for both trajectories of this wave
    const unsigned long long gin_base  = (unsigned long long)(uintptr_t)in;
    const unsigned long long gout_base = (unsigned long long)(uintptr_t)out;
    const unsigned long long gin0  = gin_base  + (unsigned long long)(2 * blockIdx.x + 0) * (TT * FF * 4ull);
    const unsigned long long gin1  = gin_base  + (unsigned long long)(2 * blockIdx.x + 1) * (TT * FF * 4ull);
    const unsigned long long gout0 = gout_base + (unsigned long long)(2 * blockIdx.x + 0) * (TT * NTT * 4ull);
    const unsigned long long gout1 = gout_base + (unsigned long long)(2 * blockIdx.x + 1) * (TT * NTT * 4ull);

    // ---- per-participant parameters (tiny, read once) ----
    const int pid = (int)in[b * (TT * FF) + (NTT + 2)];
    const float p_nvs = p_nvs_g[pid];
    const float p_nvp = p_nvp_g[pid];
    const float p_vbs = p_vbs_g[pid];
    const float p_vbp = p_vbp_g[pid];
    const float p_mp  = p_mp_g[pid];

    // ---- async-copy issue helpers (all lanes participate, no divergence) ----
    auto issue_loads = [&](int c, unsigned ldsbuf) {
        const unsigned long long coff = (unsigned long long)c * TRAJ_IN;
#pragma unroll
        for (int i = 0; i < IN_UNITS / 32; ++i) {      // 19 exact
            const int unit  = i * 32 + lane;           // 0..607
            const int tu    = (unit >= (TRAJ_IN / 16)) ? 1 : 0;
            const int local = unit - tu * (TRAJ_IN / 16);
            const unsigned long long g = (tu ? gin1 : gin0) + coff + (unsigned)(local * 16);
            const unsigned l = ldsbuf + (unsigned)(tu * TRAJ_IN + local * 16);
            async_load_b128(l, g);
        }
    };
    auto issue_stores = [&](int c) {
        const unsigned long long coff = (unsigned long long)c * TRAJ_OUT;
#pragma unroll
        for (int i = 0; i < OUT_UNITS / 32; ++i) {     // 16 exact
            const int unit  = i * 32 + lane;           // 0..511
            const int tu    = unit >> 8;               // 256 units per trajectory
            const int local = unit & 255;
            const unsigned long long g = (tu ? gout1 : gout0) + coff + (unsigned)(local * 16);
            const unsigned l = out_ldsb + (unsigned)(tu * TRAJ_OUT + local * 16);
            async_store_b128(g, l);
        }
    };

    // ---- scan state (per lane == per (trajectory, tile)) ----
    float iv_s = 0.f, iv_p = 0.f, mp = 0.f, ts_s = 0.f, ts_p = 0.f;
    float prev_t = 0.f;
    int prev_row = 0, prev_col = 0;
    const int trow = tile >> 2;
    const int tcol = tile & 3;

    // prologue: chunk 0 in flight
    issue_loads(0, in_lds0);

    for (int c = 0; c < NCHUNK; ++c) {
        // Wait for: this chunk's input loads AND the previous chunk's output
        // stores (frees the out staging buffer). Everything on ASYNCcnt.
        wait_async0();
        if (c + 1 < NCHUNK) issue_loads(c + 1, ((c + 1) & 1) ? in_lds1 : in_lds0);
        // ^ next chunk streams in while we compute below.

        // Select current input buffer at runtime (no const-initialized LDS
        // pointer arrays -- those become illegal static addrspacecast inits).
        const char* ib = (const char*)(smem + ((c & 1) ? WAVE_IN : 0) + traj * TRAJ_IN);

        for (int s = 0; s < SS; ++s) {
            const float* rec = (const float*)(ib + s * RECB);
            const float vs_f    = rec[tile];        // one-hot self entry for my tile
            const float partner = rec[NTT];         // partner tile idx (as float)
            const float timev   = rec[NTT + 1];     // time

            const bool vs_b = vs_f > 0.5f;

            // curr = argmax(one-hot) via wave32 ballot over my half-wave
            const unsigned mask = (unsigned)__ballot(vs_b);
            const unsigned half = traj ? (mask >> 16) : (mask & 0xFFFFu);
            const int curr = half ? (__ffs(half) - 1) : 0;
            const int row = curr >> 2, col = curr & 3;

            int dr = row - prev_row, dc = col - prev_col;
            if (c == 0 && s == 0) { dr = 0; dc = 0; }   // reference: first delta = 0

            const bool align = (dr * (trow - row) + dc * (tcol - col)) > 0;
            const bool vp_b  = ((int)partner == tile);
            const bool nv_b  = !vs_b && !vp_b;

            const float tp = timev * TMAXI;
            const float dt = tp - prev_t;
            ts_s += dt;
            ts_p += dt;
            if (nv_b) { iv_s += ts_s * p_nvs; iv_p += ts_p * p_nvp; }
            if (vs_b) { iv_s += ts_s * p_vbs; }
            if (vp_b) { iv_p += ts_p * p_vbp; }
            mp = align ? (mp + p_mp) : 0.f;
            if (vs_b) ts_s = 0.f;
            if (vp_b) ts_p = 0.f;

            float logit = iv_s + iv_p + mp;
            int ar = trow - row; ar = ar < 0 ? -ar : ar;
            int ac = tcol - col; ac = ac < 0 ? -ac : ac;
            logit = (ar <= 1 && ac <= 1) ? logit : OFFV;  // 8-way adjacency + stay

            prev_t = tp; prev_row = row; prev_col = col;

            out_ptr[traj * (TRAJ_OUT / 4) + s * NTT + tile] = logit;
        }

        wait_ds0();        // LDS out-stage writes visible before DMA reads them
        issue_stores(c);   // LDS -> global, overlaps with next iteration's loads
    }
    wait_async0();          // drain final stores
}

extern "C" void kernel_launch(void* const* d_in, const int* in_sizes, int n_in,
                              void* d_out, int out_size, void* d_ws, size_t ws_size,
                              hipStream_t stream) {
    (void)in_sizes; (void)n_in; (void)out_size; (void)d_ws; (void)ws_size;
    const float* in  = (const float*)d_in[0];
    const float* nvs = (const float*)d_in[1];
    const float* nvp = (const float*)d_in[2];
    const float* vbs = (const float*)d_in[3];
    const float* vbp = (const float*)d_in[4];
    const float* mpp = (const float*)d_in[5];
    float* out = (float*)d_out;

    const int B = 2048;
    dim3 grid(B / 2), block(32);   // 1 wave32 per workgroup, 2 trajectories per wave
    hipLaunchKernelGGL(ifm_scan_kernel, grid, block, 0, stream,
                       in, nvs, nvp, vbs, vbp, mpp, out);
}